// WTAGNNLayer_17849884082713
// MI455X (gfx1250) — compile-verified
//
#include <hip/hip_runtime.h>

typedef __attribute__((ext_vector_type(2))) float v2f;
typedef __attribute__((ext_vector_type(8))) float v8f;

#define N_NODES 50000
#define N_EDGES 1600000
#define FEAT 64
#define WAVES_PER_BLOCK 4
#define THREADS (WAVES_PER_BLOCK * 32)
#define A_STRIDE 132   // 16x128 LDS tile row stride (padded: 132%64=4 -> conflict-free A frags)

// -------- B-fragment staging: W[K x 64] -> LDS in v_wmma_f32_16x16x4 B layout ----------
// frag index f = ((c*4 + t)*32 + l)*2 + v ; element W[k][n] with
//   k = 4c + 2*(l>>4) + v , n = 16t + (l&15)
__device__ __forceinline__ void stage_w_frags(const float* __restrict__ W, float* lds, int K) {
  int total = K * 64;
  for (int f = threadIdx.x; f < total; f += blockDim.x) {
    int v = f & 1;
    int l = (f >> 1) & 31;
    int t = (f >> 6) & 3;
    int c = f >> 8;
    int k = 4 * c + 2 * (l >> 4) + v;
    int n = 16 * t + (l & 15);
    lds[f] = W[k * 64 + n];
  }
}

// -------- 16x64 = 16x64 @ 64x64 GEMM core (A row-major in global, B frags in LDS) -------
__device__ __forceinline__ void gemm16x64_k64(const float* __restrict__ Arows,
                                              const float* Bfrag, int lane, v8f acc[4]) {
  const float* ar = Arows + (size_t)(lane & 15) * 64 + 2 * (lane >> 4);
#pragma unroll
  for (int c = 0; c < 16; ++c) {
    v2f a = *(const v2f*)(ar + 4 * c);
#pragma unroll
    for (int t = 0; t < 4; ++t) {
      v2f b = *(const v2f*)(Bfrag + ((c * 4 + t) * 32 + lane) * 2);
      acc[t] = __builtin_amdgcn_wmma_f32_16x16x4_f32(false, a, false, b, (short)0,
                                                     acc[t], false, false);
    }
  }
}

// ------------------------------ kernel 0: zero scratch ---------------------------------
__global__ void zero_kernel(float* __restrict__ p, int n) {
  int i = blockIdx.x * blockDim.x + threadIdx.x;
  if (i < n) p[i] = 0.0f;
}

// -------------------- kernel 1: nf_out = relu(nf @ W_node + b_node) --------------------
__global__ __launch_bounds__(THREADS) void node_kernel(const float* __restrict__ nf,
                                                       const float* __restrict__ W_node,
                                                       const float* __restrict__ b_node,
                                                       float* __restrict__ nf_out) {
  __shared__ float Bfrag[64 * 64];
  stage_w_frags(W_node, Bfrag, 64);
  __syncthreads();
  int wave = blockIdx.x * WAVES_PER_BLOCK + (threadIdx.x >> 5);
  int lane = threadIdx.x & 31;
  if (wave >= N_NODES / 16) return;          // wave-uniform: EXEC all-ones inside
  int row0 = wave * 16;
  v8f acc[4] = {};
  gemm16x64_k64(nf + (size_t)row0 * 64, Bfrag, lane, acc);
  int colbase = lane & 15;
  int mhalf = (lane >> 4) * 8;
#pragma unroll
  for (int t = 0; t < 4; ++t) {
    float bb = b_node[t * 16 + colbase];
#pragma unroll
    for (int r = 0; r < 8; ++r) {
      float v = fmaxf(acc[t][r] + bb, 0.0f);
      nf_out[(size_t)(row0 + mhalf + r) * 64 + t * 16 + colbase] = v;
    }
  }
}

// ---------- kernel 2: ef_p = ef @ W_edge ; scatter-add into sums[dst], cnt[dst] --------
__global__ __launch_bounds__(THREADS) void edge_proj_kernel(const float* __restrict__ ef,
                                                            const float* __restrict__ W_edge,
                                                            const int* __restrict__ dst,
                                                            float* __restrict__ sums,
                                                            float* __restrict__ cnt) {
  __shared__ float Bfrag[64 * 64];
  stage_w_frags(W_edge, Bfrag, 64);
  __syncthreads();
  int wave = blockIdx.x * WAVES_PER_BLOCK + (threadIdx.x >> 5);
  int lane = threadIdx.x & 31;
  int e0 = wave * 16;                         // grid covers edges exactly
  v8f acc[4] = {};
  gemm16x64_k64(ef + (size_t)e0 * 64, Bfrag, lane, acc);
  int colbase = lane & 15;
  int mhalf = (lane >> 4) * 8;
#pragma unroll
  for (int r = 0; r < 8; ++r) {
    int d = dst[e0 + mhalf + r];
#pragma unroll
    for (int t = 0; t < 4; ++t)
      atomicAdd(&sums[(size_t)d * 64 + t * 16 + colbase], acc[t][r]);
  }
  if (lane < 16) atomicAdd(&cnt[dst[e0 + lane]], 1.0f);
}

// --------------------- kernel 3: nb = sums / max(cnt, 1) (in place) --------------------
__global__ void finalize_kernel(float* __restrict__ sums, const float* __restrict__ cnt) {
  int i = blockIdx.x * blockDim.x + threadIdx.x;
  if (i < N_NODES * 64) {
    float c = cnt[i >> 6];
    sums[i] = sums[i] / fmaxf(c, 1.0f);
  }
}

// ------ kernel 4: fused edge output: recompute ef_p, gather, K=128 GEMM, bias+relu -----
__global__ __launch_bounds__(THREADS) void edge_out_kernel(
    const float* __restrict__ ef, const float* __restrict__ W_edge,
    const float* __restrict__ W_dense, const float* __restrict__ b_edge,
    const float* __restrict__ b_dense, const int* __restrict__ src,
    const int* __restrict__ dst, const float* __restrict__ nb,
    const float* __restrict__ nf_out, float* __restrict__ ef_out) {
  __shared__ float BfragE[64 * 64];                       // 16 KB
  __shared__ float BfragD[128 * 64];                      // 32 KB
  __shared__ float Atile[WAVES_PER_BLOCK][16 * A_STRIDE]; // 33 KB
  stage_w_frags(W_edge, BfragE, 64);
  stage_w_frags(W_dense, BfragD, 128);
  __syncthreads();

  int waveInBlk = threadIdx.x >> 5;
  int wave = blockIdx.x * WAVES_PER_BLOCK + waveInBlk;
  int lane = threadIdx.x & 31;
  int e0 = wave * 16;
  float* At = &Atile[waveInBlk][0];
  int colbase = lane & 15;
  int mhalf = (lane >> 4) * 8;

  // GEMM 1: ef_p tile (16x64)
  v8f acc[4] = {};
  gemm16x64_k64(ef + (size_t)e0 * 64, BfragE, lane, acc);

  // tmp_ef = ef_p + nb[dst]  -> LDS cols 0..63 (C-layout unpack fused with gather)
#pragma unroll
  for (int r = 0; r < 8; ++r) {
    int m = mhalf + r;
    int d = dst[e0 + m];
#pragma unroll
    for (int t = 0; t < 4; ++t) {
      At[m * A_STRIDE + t * 16 + colbase] =
          acc[t][r] + nb[(size_t)d * 64 + t * 16 + colbase];
    }
  }

  // srcdst = 0.5*(nf_out[src]+nf_out[dst]) -> LDS cols 64..127 (2 cols per lane)
#pragma unroll 4
  for (int e = 0; e < 16; ++e) {
    int s = src[e0 + e], d = dst[e0 + e];
    v2f a = *(const v2f*)&nf_out[(size_t)s * 64 + 2 * lane];
    v2f b = *(const v2f*)&nf_out[(size_t)d * 64 + 2 * lane];
    v2f m = (a + b) * 0.5f;
    *(v2f*)&At[e * A_STRIDE + 64 + 2 * lane] = m;
  }

  // GEMM 2: [16x128] @ [128x64], A frags from LDS (padded stride -> no bank conflicts)
  v8f acc2[4] = {};
  {
    const float* ar = At + (lane & 15) * A_STRIDE + 2 * (lane >> 4);
#pragma unroll
    for (int c = 0; c < 32; ++c) {
      v2f a = *(const v2f*)(ar + 4 * c);
#pragma unroll
      for (int t = 0; t < 4; ++t) {
        v2f b = *(const v2f*)(BfragD + ((c * 4 + t) * 32 + lane) * 2);
        acc2[t] = __builtin_amdgcn_wmma_f32_16x16x4_f32(false, a, false, b, (short)0,
                                                        acc2[t], false, false);
      }
    }
  }

  // epilogue: + b_dense + b_edge, relu, store
#pragma unroll
  for (int t = 0; t < 4; ++t) {
    int col = t * 16 + colbase;
    float bb = b_dense[col] + b_edge[col];
#pragma unroll
    for (int r = 0; r < 8; ++r) {
      float v = fmaxf(acc2[t][r] + bb, 0.0f);
      ef_out[(size_t)(e0 + mhalf + r) * 64 + col] = v;
    }
  }
}

extern "C" void kernel_launch(void* const* d_in, const int* in_sizes, int n_in,
                              void* d_out, int out_size, void* d_ws, size_t ws_size,
                              hipStream_t stream) {
  const float* nf      = (const float*)d_in[0];
  const float* ef      = (const float*)d_in[1];
  const int*   src     = (const int*)d_in[2];
  const int*   dst     = (const int*)d_in[3];
  const float* W_node  = (const float*)d_in[4];
  const float* W_edge  = (const float*)d_in[5];
  const float* b_node  = (const float*)d_in[6];
  const float* b_edge  = (const float*)d_in[7];
  const float* W_dense = (const float*)d_in[8];
  const float* b_dense = (const float*)d_in[9];

  float* nf_out = (float*)d_out;
  float* ef_out = nf_out + (size_t)N_NODES * 64;
  float* sums   = (float*)d_ws;                      // 50000*64 floats
  float* cnt    = sums + (size_t)N_NODES * 64;       // 50000 floats

  int nzero = N_NODES * 64 + N_NODES;
  zero_kernel<<<(nzero + 255) / 256, 256, 0, stream>>>(sums, nzero);

  int node_waves = N_NODES / 16;                     // 3125
  node_kernel<<<(node_waves + WAVES_PER_BLOCK - 1) / WAVES_PER_BLOCK, THREADS, 0, stream>>>(
      nf, W_node, b_node, nf_out);

  int edge_blocks = (N_EDGES / 16) / WAVES_PER_BLOCK; // 25000 exact
  edge_proj_kernel<<<edge_blocks, THREADS, 0, stream>>>(ef, W_edge, dst, sums, cnt);

  finalize_kernel<<<(N_NODES * 64 + 255) / 256, 256, 0, stream>>>(sums, cnt);

  edge_out_kernel<<<edge_blocks, THREADS, 0, stream>>>(
      ef, W_edge, W_dense, b_edge, b_dense, src, dst, sums, nf_out, ef_out);
}